// E2TensorProductArbitraryOrder_woequal_89979564851583
// MI455X (gfx1250) — compile-verified
//
#include <hip/hip_runtime.h>

typedef float v4f __attribute__((ext_vector_type(4)));

// Real-basis Wigner constants (derived analytically from the reference's basis change)
#define K3  0.5773502691896258f   // 1/sqrt(3)
#define R2  0.7071067811865476f   // 1/sqrt(2)
#define R6  0.4082482904638630f   // 1/sqrt(6)
#define K5  0.4472135954999579f   // 1/sqrt(5)
#define K30 0.18257418583505536f  // 1/sqrt(30)
#define S3  1.7320508075688772f   // sqrt(3)

// Issue one sample's x1 row-block (9 rows x 128 floats; this lane's 16B of each row)
// as CDNA5 async global->LDS copies (ASYNCcnt-tracked, no VGPR data staging).
// x1 is streamed exactly once -> non-temporal hint keeps it out of L2.
__device__ __forceinline__ void issue_z(const float* __restrict__ x1, int z,
                                        unsigned ldsBufB, unsigned laneB) {
  unsigned long long g = (unsigned long long)(const void*)x1
                       + (unsigned long long)z * 4608ull + (unsigned long long)laneB;
#pragma unroll
  for (int i = 0; i < 9; ++i) {
    unsigned lds_addr = ldsBufB + (unsigned)i * 512u + laneB;
    asm volatile("global_load_async_to_lds_b128 %0, %1, off th:TH_LOAD_NT"
                 :: "v"(lds_addr), "v"(g)
                 : "memory");
    g += 512ull;
  }
}

__global__ __launch_bounds__(256) void e2tp_kernel(
    const float* __restrict__ x1, const float* __restrict__ x2,
    const float* __restrict__ w,  float* __restrict__ out, int N) {
  // 8 waves * 2 buffers * 9 rows * 128 floats = 18432 floats = 72 KB
  __shared__ __align__(16) float smem[18432];

  const int lane = threadIdx.x & 31;
  const int wib  = threadIdx.x >> 5;          // wave in block
  const int wpb  = blockDim.x >> 5;
  const int gwave  = blockIdx.x * wpb + wib;
  const int nwaves = gridDim.x * wpb;

  // Absolute LDS byte offset of smem (AS(3) pointers are the raw LDS offset).
  unsigned smem_base;
  {
    __attribute__((address_space(3))) const float* sp3 =
        (__attribute__((address_space(3))) const float*)smem;
    smem_base = (unsigned)(unsigned long long)sp3;
  }
  const unsigned laneB = (unsigned)lane * 16u;
  const unsigned waveB = smem_base + (unsigned)wib * 9216u;
  const int sBaseF = wib * 2304 + lane * 4;   // float-index base for LDS reads

  // Per-lane weights are invariant over z: hoist out of the loop (RT -> cached).
  float wq[6][4];
#pragma unroll
  for (int p = 0; p < 6; ++p) {
    const v4f t = *(const v4f*)(w + p * 128 + lane * 4);
    wq[p][0] = t.x; wq[p][1] = t.y; wq[p][2] = t.z; wq[p][3] = t.w;
  }

  int z = gwave;
  if (z >= N) return;

  issue_z(x1, z, waveB, laneB);
  int buf = 0;

  while (true) {
    const int  zn   = z + nwaves;
    const bool more = (zn < N);
    if (more) {
      issue_z(x1, zn, waveB + (unsigned)(buf ^ 1) * 4608u, laneB);
      // 9 loads just issued for zn; async loads retire in order, so <=9
      // outstanding means all 9 for the current z have landed in LDS.
      asm volatile("s_wait_asynccnt 0x9" ::: "memory");
    } else {
      asm volatile("s_wait_asynccnt 0x0" ::: "memory");
    }

    // Pull the staged tile out of LDS (9 x ds_load_b128 per lane).
    float X[9][4];
#pragma unroll
    for (int i = 0; i < 9; ++i) {
      const v4f t = *(const v4f*)(&smem[sBaseF + buf * 1152 + i * 128]);
      X[i][0] = t.x; X[i][1] = t.y; X[i][2] = t.z; X[i][3] = t.w;
    }
    const float v0 = x2[3 * z + 0];
    const float v1 = x2[3 * z + 1];
    const float v2 = x2[3 * z + 2];

    float O[9][4];
#pragma unroll
    for (int c = 0; c < 4; ++c) {
      const float s  = X[0][c];
      const float p0 = X[1][c], p1 = X[2][c], p2 = X[3][c];
      const float d0 = X[4][c], d1 = X[5][c], d2 = X[6][c], d3 = X[7][c], d4 = X[8][c];
      const float w0 = wq[0][c], w1 = wq[1][c], w2 = wq[2][c];
      const float w3 = wq[3][c], w4 = wq[4][c], w5 = wq[5][c];

      // path (1,1,0): delta/sqrt3
      const float dot = p0 * v0 + p1 * v1 + p2 * v2;
      O[0][c] = w1 * (K3 * dot);

      // path (1,1,1): epsilon/sqrt6 (cross product in (y,z,x) order)
      const float c0 = p1 * v2 - p2 * v1;
      const float c1 = p2 * v0 - p0 * v2;
      const float c2 = p0 * v1 - p1 * v0;

      // path (2,1,1): (D v)/sqrt5, D = sum_a d_a M_a
      const float t0 = R2 * (d0 * v2 + d1 * v1) - (R6 * d2 + R2 * d4) * v0;
      const float t1 = R2 * (d1 * v0 + d3 * v2) + (2.0f * R6) * d2 * v1;
      const float t2 = R2 * (d0 * v0 + d3 * v1) + (R2 * d4 - R6 * d2) * v2;

      const float a0 = w0 * (K3 * s);  // path (0,1,1)
      O[1][c] = a0 * v0 + w2 * (R6 * c0) + w4 * (K5 * t0);
      O[2][c] = a0 * v1 + w2 * (R6 * c1) + w4 * (K5 * t1);
      O[3][c] = a0 * v2 + w2 * (R6 * c2) + w4 * (K5 * t2);

      // path (1,1,2): <M_a, p (x) v>/sqrt5
      const float e0 = R2 * (p2 * v0 + p0 * v2);
      const float e1 = R2 * (p0 * v1 + p1 * v0);
      const float e2 = R6 * (2.0f * p1 * v1 - p2 * v2 - p0 * v0);
      const float e3 = R2 * (p1 * v2 + p2 * v1);
      const float e4 = R2 * (p2 * v2 - p0 * v0);

      // path (2,1,2): adjoint action Tr(M_b [L_c, M_a]) d_a v_c / sqrt30
      const float g0 = d1 * v0 + 2.0f * d4 * v1 - d3 * v2;
      const float g1 = -d0 * v0 + d3 * v1 - (S3 * d2 + d4) * v2;
      const float g2 = S3 * (d1 * v2 - d3 * v0);
      const float g3 = (S3 * d2 - d4) * v0 - d1 * v1 + d0 * v2;
      const float g4 = d3 * v0 - 2.0f * d0 * v1 + d1 * v2;

      const float w3s = w3 * K5;
      const float w5s = w5 * K30;
      O[4][c] = w3s * e0 + w5s * g0;
      O[5][c] = w3s * e1 + w5s * g1;
      O[6][c] = w3s * e2 + w5s * g2;
      O[7][c] = w3s * e3 + w5s * g3;
      O[8][c] = w3s * e4 + w5s * g4;
    }

    // Output is written once and never re-read: non-temporal B128 stores.
    float* ob = out + (size_t)z * 1152 + lane * 4;
#pragma unroll
    for (int k = 0; k < 9; ++k) {
      v4f t;
      t.x = O[k][0]; t.y = O[k][1]; t.z = O[k][2]; t.w = O[k][3];
      __builtin_nontemporal_store(t, (v4f*)(ob + k * 128));
    }

    if (!more) break;
    z = zn;
    buf ^= 1;
  }
}

extern "C" void kernel_launch(void* const* d_in, const int* in_sizes, int n_in,
                              void* d_out, int out_size, void* d_ws, size_t ws_size,
                              hipStream_t stream) {
  (void)n_in; (void)out_size; (void)d_ws; (void)ws_size;
  const float* x1 = (const float*)d_in[0];
  const float* x2 = (const float*)d_in[1];
  const float* w  = (const float*)d_in[2];
  float* out = (float*)d_out;

  const int N = in_sizes[1] / 3;   // x2 is (N, 3, 1)
  const int block = 256;           // 8 wave32 per block
  const int wpb = block / 32;
  int blocks = 1024;               // 8192 waves -> ~12 samples each (pipeline depth)
  if (blocks * wpb > N) blocks = (N + wpb - 1) / wpb;
  if (blocks < 1) blocks = 1;

  e2tp_kernel<<<blocks, block, 0, stream>>>(x1, x2, w, out, N);
}